// HyperModel_47115791237245
// MI455X (gfx1250) — compile-verified
//
#include <hip/hip_runtime.h>
#include <hip/hip_bf16.h>

// MI455X HyperLSTM forward.
// 64 blocks x 256 threads (8 wave32 waves). Block b owns batch rows
// [16b,16b+16) and runs the full T=512 x 2-layer recurrence with ALL weights
// (~244KB padded) + states + activations resident in the WGP's 320KB LDS
// (295.6KB used). Every matmul is v_wmma_f32_16x16x4_f32 (fp32 in/out =>
// reference-exact arithmetic). Weight B-operands are either LDS-padded
// (unconditional ds_loads) or clamped-address + cndmask (no exec-mask dances,
// loads stay batchable). rec/inp/d_h/d_x/d_b are fused per (group, half-tile)
// wave job and combined in registers. ~58 GFLOP total, ~42MB HBM traffic:
// purely latency-bound on the sequential scan, so minimize per-step latency.

typedef __attribute__((ext_vector_type(2))) float v2f;
typedef __attribute__((ext_vector_type(8))) float v8f;

#define NTH 256

constexpr int Bc = 1024, Tc = 512, INc = 11, HIDc = 25, HYPc = 26, NZc = 24, MAPc = 9;
constexpr float EPSc = 1e-5f;

// ---- shared weights (padded) ----
constexpr int O_WE  = 0;      // We  32 x 12  (valid 25 x 11)
constexpr int O_BE  = 384;    // be  32       (valid 25)
constexpr int O_WR  = 416;    // Wr  16 x 28  (valid 9 x 25)
constexpr int O_BR  = 864;    // br  16       (valid 9)
constexpr int O_LAY = 880;

// ---- per-layer weight sub-offsets ----
constexpr int W_HYPX = 0;      // Whyp_x 112 x 52 (valid 104 x 50)
constexpr int W_HYPH = 5824;   // Whyp_h 112 x 28 (valid 104 x 26)
constexpr int W_BHYP = 8960;   // bhyp 112 (valid 104)
constexpr int W_LHG  = 9072;   // ln_hyp_g 104
constexpr int W_LHB  = 9176;   // ln_hyp_b 104
constexpr int W_LHCG = 9280;   // ln_hypc_g 26
constexpr int W_LHCB = 9306;   // ln_hypc_b 26
constexpr int W_Z    = 9332;   // [Wzh;Wzx;Wzb] 288 x 28 (valid K 26)
constexpr int W_BZ   = 17396;  // [bzh;bzx;0] 288
constexpr int W_DH   = 17684;  // Wdh 4 x 25 x 24 (tight, n-guarded)
constexpr int W_DX   = 20084;
constexpr int W_DB   = 22484;
constexpr int W_BDB  = 24884;  // bdb 100
constexpr int W_WH   = 24984;  // w_h 4 x 25 x 28 (valid K 25, n-guarded)
constexpr int W_WX   = 27784;  // w_x 4 x 25 x 28
constexpr int W_LG   = 30584;  // ln_g 100
constexpr int W_LB   = 30684;  // ln_b 100
constexpr int W_LCG  = 30784;  // ln_c_g 25
constexpr int W_LCB  = 30809;  // ln_c_b 25
constexpr int W_LAYER_SZ = 30834;

constexpr int O_ACT = O_LAY + 2 * W_LAYER_SZ;  // 62548

// ---- activation/state sub-offsets (relative to O_ACT) ----
constexpr int A_XT   = 0;      // 16 x 12
constexpr int A_XIN  = 192;    // 16 x 32
constexpr int A_XH   = 704;    // 16 x 52
constexpr int A_H    = 1536;   // 2 x 16 x 28
constexpr int A_C    = 2432;   // 2 x 16 x 25
constexpr int A_HH   = 3232;   // 2 x 16 x 28
constexpr int A_CH   = 4128;   // 2 x 16 x 26
constexpr int A_IFGO = 4960;   // 16 x 112
constexpr int A_Z    = 6752;   // 16 x 288
constexpr int A_GATE = 11360;  // 16 x 100
constexpr int A_RST  = 12960;  // 16 x 2
constexpr int A_NUM  = 12992;  // 16 x 9
constexpr int A_SZ   = 13136;

constexpr int    LDS_FLOATS = O_ACT + A_SZ;            // 75684
constexpr size_t LDS_BYTES  = (size_t)LDS_FLOATS * 4;  // 302736 B
static_assert(LDS_BYTES <= 320 * 1024, "fits WGP LDS");

// output offsets (return order: num, m, h, c, hh, ch)
constexpr size_t OUT_M  = (size_t)Bc * MAPc;
constexpr size_t OUT_H  = OUT_M + (size_t)Bc * Tc * MAPc;
constexpr size_t OUT_C  = OUT_H + 2ull * Bc * HIDc;
constexpr size_t OUT_HH = OUT_C + 2ull * Bc * HIDc;
constexpr size_t OUT_CH = OUT_HH + 2ull * Bc * HYPc;

struct Args { const float* q[53]; };

__device__ __forceinline__ float sigm(float x) { return 1.0f / (1.0f + __expf(-x)); }

__device__ __forceinline__ v8f v8zero() {
    v8f z;
    for (int i = 0; i < 8; ++i) z[i] = 0.f;
    return z;
}

// 16x16 tile: acc += A(16 x Kp, LDS, zero-padded) * B, B[k][n] = W[n*Kp+k],
// weights LDS-padded so every load is unconditional.
__device__ __forceinline__ v8f wmma_np(const float* __restrict__ A, int lda,
                                       const float* __restrict__ W, int Kp,
                                       int n0, v8f acc, int lane) {
    const int half2 = (lane >> 4) << 1;  // 0 or 2
    const int r     = lane & 15;
    const float* Wn = W + (n0 + r) * Kp;
#pragma unroll
    for (int k = 0; k < 128; k += 4) {
        if (k >= Kp) break;
        const int ka = k + half2;
        v2f a; a.x = A[r * lda + ka]; a.y = A[r * lda + ka + 1];
        v2f b; b.x = Wn[ka];          b.y = Wn[ka + 1];
        acc = __builtin_amdgcn_wmma_f32_16x16x4_f32(false, a, false, b,
                                                    (short)0, acc, false, false);
    }
    return acc;
}

// Same, but rows beyond N masked via clamped address + select (no exec dance).
__device__ __forceinline__ v8f wmma_ng(const float* __restrict__ A, int lda,
                                       const float* __restrict__ W, int N, int Kp,
                                       int n0, v8f acc, int lane) {
    const int half2 = (lane >> 4) << 1;
    const int r     = lane & 15;
    const int n     = n0 + r;
    const bool nok  = (n < N);
    const float* Wn = W + (nok ? n : 0) * Kp;   // always a safe address
    const float msk = nok ? 1.f : 0.f;
#pragma unroll
    for (int k = 0; k < 128; k += 4) {
        if (k >= Kp) break;
        const int ka = k + half2;
        v2f a; a.x = A[r * lda + ka];  a.y = A[r * lda + ka + 1];
        v2f b; b.x = Wn[ka] * msk;     b.y = Wn[ka + 1] * msk;
        acc = __builtin_amdgcn_wmma_f32_16x16x4_f32(false, a, false, b,
                                                    (short)0, acc, false, false);
    }
    return acc;
}

// Store D tile + bias[n] (bias arrays padded to tile width).
__device__ __forceinline__ void store_tile(float* __restrict__ Out, int ldo, int n0,
                                           v8f d, const float* bias, int lane) {
    const int col = lane & 15;
    const int rb  = (lane >> 4) << 3;
    const int n   = n0 + col;
    const float bv = bias ? bias[n] : 0.f;
#pragma unroll
    for (int i = 0; i < 8; ++i) Out[(rb + i) * ldo + n] = d[i] + bv;
}

__global__ __launch_bounds__(NTH, 1) void hyperlstm_kernel(Args a, float* __restrict__ out) {
    extern __shared__ float S[];
    const int tid  = threadIdx.x;
    const int lane = tid & 31;
    const int wave = tid >> 5;
    const int b0   = blockIdx.x * 16;

    // ---------- zero ALL of LDS (pads stay zero forever) ----------
    for (int i = tid; i < LDS_FLOATS; i += NTH) S[i] = 0.f;
    __syncthreads();

    // ---------- stage weights + states ----------
    auto cpy = [&](int dst, const float* src, int n) {
        for (int i = tid; i < n; i += NTH) S[dst + i] = src[i];
    };
    auto cpy2d = [&](int dst, const float* src, int rows, int cols, int ldd) {
        for (int i = tid; i < rows * cols; i += NTH) {
            const int r = i / cols, c = i % cols;
            S[dst + r * ldd + c] = src[i];
        }
    };
    cpy2d(O_WE, a.q[5], 25, 11, 12);  cpy(O_BE, a.q[7], 25);
    cpy2d(O_WR, a.q[6], 9, 25, 28);   cpy(O_BR, a.q[8], 9);
    for (int l = 0; l < 2; ++l) {
        const float* const* q = a.q + 9 + l * 22;  // sorted-key layer leaves
        const int W = O_LAY + l * W_LAYER_SZ;
        cpy2d(W + W_HYPX, q[4], 104, 50, 52);      // Whyp_x
        cpy2d(W + W_HYPH, q[3], 104, 26, 28);      // Whyp_h
        cpy(W + W_BHYP, q[9], 104);
        cpy(W + W_LHG,  q[17], 104);
        cpy(W + W_LHB,  q[16], 104);
        cpy(W + W_LHCG, q[19], 26);
        cpy(W + W_LHCB, q[18], 26);
        cpy2d(W + W_Z,            q[6], 96, 26, 28);   // Wzh
        cpy2d(W + W_Z +  96 * 28, q[7], 96, 26, 28);   // Wzx
        cpy2d(W + W_Z + 192 * 28, q[5], 96, 26, 28);   // Wzb
        cpy(W + W_BZ,      q[10], 96);                 // bzh
        cpy(W + W_BZ + 96, q[11], 96);                 // bzx (Wzb bias = 0 from memset)
        cpy(W + W_DH, q[1], 2400);
        cpy(W + W_DX, q[2], 2400);
        cpy(W + W_DB, q[0], 2400);
        cpy(W + W_BDB, q[8], 100);
        cpy2d(W + W_WH, q[20], 100, 25, 28);       // w_h (row = g*25+k)
        cpy2d(W + W_WX, q[21], 100, 25, 28);       // w_x
        cpy(W + W_LG,  q[15], 100);
        cpy(W + W_LB,  q[12], 100);
        cpy(W + W_LCG, q[14], 25);
        cpy(W + W_LCB, q[13], 25);
    }
    for (int e = tid; e < 2 * 16 * HIDc; e += NTH) {
        const int l = e / 400, rem = e % 400, r = rem / HIDc, k = rem % HIDc;
        const size_t gi = (size_t)l * Bc * HIDc + (size_t)(b0 + r) * HIDc + k;
        S[O_ACT + A_H + l * 448 + r * 28 + k]   = a.q[1][gi];
        S[O_ACT + A_C + l * 400 + r * HIDc + k] = a.q[2][gi];
    }
    for (int e = tid; e < 2 * 16 * HYPc; e += NTH) {
        const int l = e / 416, rem = e % 416, r = rem / HYPc, j = rem % HYPc;
        const size_t gi = (size_t)l * Bc * HYPc + (size_t)(b0 + r) * HYPc + j;
        S[O_ACT + A_HH + l * 448 + r * 28 + j]   = a.q[3][gi];
        S[O_ACT + A_CH + l * 416 + r * HYPc + j] = a.q[4][gi];
    }
    __syncthreads();

    float* const xt   = S + O_ACT + A_XT;
    float* const xin  = S + O_ACT + A_XIN;
    float* const xh   = S + O_ACT + A_XH;
    float* const ifgo = S + O_ACT + A_IFGO;
    float* const zb   = S + O_ACT + A_Z;
    float* const gat  = S + O_ACT + A_GATE;
    float* const rst  = S + O_ACT + A_RST;
    float* const nac  = S + O_ACT + A_NUM;

    for (int t = 0; t < Tc; ++t) {
        for (int l = 0; l < 2; ++l) {
            const float* W = S + O_LAY + l * W_LAYER_SZ;
            float* hS  = S + O_ACT + A_H  + l * 448;
            float* cS  = S + O_ACT + A_C  + l * 400;
            float* hhS = S + O_ACT + A_HH + l * 448;
            float* chS = S + O_ACT + A_CH + l * 416;

            // ---- P0: layer input -> xin ----
            if (l == 0) {
                for (int e = tid; e < 16 * INc; e += NTH) {
                    const int r = e / INc, j = e % INc;
                    xt[r * 12 + j] = a.q[0][((size_t)(b0 + r) * Tc + t) * INc + j];
                }
                if (t + 1 < Tc && tid < 16)  // prefetch next step's x rows
                    __builtin_prefetch(a.q[0] + ((size_t)(b0 + tid) * Tc + (t + 1)) * INc, 0, 1);
                __syncthreads();
                for (int j = wave; j < 2; j += 8) {   // xe = leaky_relu(x@We^T+be)
                    v8f acc = wmma_np(xt, 12, S + O_WE, 12, j * 16, v8zero(), lane);
                    const int col = lane & 15, rb = (lane >> 4) << 3, n = j * 16 + col;
                    const float bv = S[O_BE + n];
#pragma unroll
                    for (int i = 0; i < 8; ++i) {
                        const float v = acc[i] + bv;
                        xin[(rb + i) * 32 + n] = (v > 0.f) ? v : 0.1f * v;
                    }
                }
            } else {
                const float* h0S = S + O_ACT + A_H;   // layer-0 h (fresh this step)
                for (int e = tid; e < 16 * HIDc; e += NTH) {
                    const int r = e / HIDc, k = e % HIDc;
                    xin[r * 32 + k] = h0S[r * 28 + k];
                }
            }
            __syncthreads();

            // ---- P1: xh = [h | xin] ----
            for (int e = tid; e < 16 * 50; e += NTH) {
                const int r = e / 50, k = e % 50;
                xh[r * 52 + k] = (k < HIDc) ? hS[r * 28 + k] : xin[r * 32 + (k - HIDc)];
            }
            __syncthreads();

            // ---- P2: ifgo = xh@Whyp_x^T + hh@Whyp_h^T + bhyp (7 tiles) ----
            for (int j = wave; j < 7; j += 8) {
                v8f acc = v8zero();
                acc = wmma_np(xh, 52, W + W_HYPX, 52, j * 16, acc, lane);
                acc = wmma_np(hhS, 28, W + W_HYPH, 28, j * 16, acc, lane);
                store_tile(ifgo, 112, j * 16, acc, W + W_BHYP, lane);
            }
            __syncthreads();

            // ---- P3a: hyper group LayerNorm ----
            if (tid < 64) {
                const int r = tid >> 2, g = tid & 3;
                float* p = ifgo + r * 112 + g * HYPc;
                const float* gw = W + W_LHG + g * HYPc;
                const float* gb = W + W_LHB + g * HYPc;
                float m = 0.f;
                for (int j = 0; j < HYPc; ++j) m += p[j];
                m *= (1.f / HYPc);
                float v = 0.f;
                for (int j = 0; j < HYPc; ++j) { const float d = p[j] - m; v += d * d; }
                v *= (1.f / HYPc);
                const float rs = rsqrtf(v + EPSc);
                for (int j = 0; j < HYPc; ++j) p[j] = (p[j] - m) * rs * gw[j] + gb[j];
            }
            __syncthreads();
            // ---- P3b: ch update ----
            for (int e = tid; e < 16 * HYPc; e += NTH) {
                const int r = e / HYPc, j = e % HYPc;
                const float hi = ifgo[r * 112 + j];
                const float hf = ifgo[r * 112 + HYPc + j];
                const float hg = ifgo[r * 112 + 2 * HYPc + j];
                chS[r * HYPc + j] = sigm(hf) * chS[r * HYPc + j] + sigm(hi) * tanhf(hg);
            }
            __syncthreads();
            // ---- P3c: row LN stats over ch ----
            if (tid < 16) {
                const float* p = chS + tid * HYPc;
                float m = 0.f;
                for (int j = 0; j < HYPc; ++j) m += p[j];
                m *= (1.f / HYPc);
                float v = 0.f;
                for (int j = 0; j < HYPc; ++j) { const float d = p[j] - m; v += d * d; }
                v *= (1.f / HYPc);
                rst[tid * 2] = m; rst[tid * 2 + 1] = rsqrtf(v + EPSc);
            }
            __syncthreads();
            // ---- P3d: hh = sig(ho)*tanh(LN(ch)) ----
            for (int e = tid; e < 16 * HYPc; e += NTH) {
                const int r = e / HYPc, j = e % HYPc;
                const float ho = ifgo[r * 112 + 3 * HYPc + j];
                const float nrm = (chS[r * HYPc + j] - rst[r * 2]) * rst[r * 2 + 1]
                                  * W[W_LHCG + j] + W[W_LHCB + j];
                hhS[r * 28 + j] = sigm(ho) * tanhf(nrm);
            }
            __syncthreads();

            // ---- P4: z = hh @ [Wzh;Wzx;Wzb]^T + [bzh;bzx;0] (18 tiles) ----
            for (int j = wave; j < 18; j += 8) {
                v8f acc = wmma_np(hhS, 28, W + W_Z, 28, j * 16, v8zero(), lane);
                store_tile(zb, 288, j * 16, acc, W + W_BZ, lane);
            }
            __syncthreads();

            // ---- P5: fused rec/inp/d_h/d_x/d_b per (group, half-tile) wave job ----
            {
                const int g = wave >> 1, t2 = wave & 1;
                v8f vr = wmma_ng(hS,  28, W + W_WH + g * 700, HIDc, 28, t2 * 16, v8zero(), lane);
                v8f vi = wmma_ng(xin, 32, W + W_WX + g * 700, HIDc, 28, t2 * 16, v8zero(), lane);
                v8f ah = wmma_ng(zb + g * NZc,       288, W + W_DH + g * 600, HIDc, 24, t2 * 16, v8zero(), lane);
                v8f ax = wmma_ng(zb + 96 + g * NZc,  288, W + W_DX + g * 600, HIDc, 24, t2 * 16, v8zero(), lane);
                v8f ab = wmma_ng(zb + 192 + g * NZc, 288, W + W_DB + g * 600, HIDc, 24, t2 * 16, v8zero(), lane);
                const int col = lane & 15, rb = (lane >> 4) << 3;
                const int kk = t2 * 16 + col;
                if (kk < HIDc) {
                    const int gidx = g * HIDc + kk;
                    const float bd = W[W_BDB + gidx];
#pragma unroll
                    for (int i = 0; i < 8; ++i)
                        gat[(rb + i) * 100 + gidx] = ah[i] * vr[i] + ax[i] * vi[i] + ab[i] + bd;
                }
            }
            __syncthreads();

            // ---- P6a: main group LayerNorm on gates ----
            if (tid < 64) {
                const int r = tid >> 2, g = tid & 3;
                float* p = gat + r * 100 + g * HIDc;
                const float* gw = W + W_LG + g * HIDc;
                const float* gb = W + W_LB + g * HIDc;
                float m = 0.f;
                for (int j = 0; j < HIDc; ++j) m += p[j];
                m *= (1.f / HIDc);
                float v = 0.f;
                for (int j = 0; j < HIDc; ++j) { const float d = p[j] - m; v += d * d; }
                v *= (1.f / HIDc);
                const float rs = rsqrtf(v + EPSc);
                for (int j = 0; j < HIDc; ++j) p[j] = (p[j] - m) * rs * gw[j] + gb[j];
            }
            __syncthreads();
            // ---- P6b: c update ----
            for (int e = tid; e < 16 * HIDc; e += NTH) {
                const int r = e / HIDc, k = e % HIDc;
                const float gi = gat[r * 100 + k];
                const float gf = gat[r * 100 + HIDc + k];
                const float gg = gat[r * 100 + 2 * HIDc + k];
                cS[r * HIDc + k] = sigm(gf) * cS[r * HIDc + k] + sigm(gi) * tanhf(gg);
            }
            __syncthreads();
            // ---- P6c: row LN stats over c ----
            if (tid < 16) {
                const float* p = cS + tid * HIDc;
                float m = 0.f;
                for (int j = 0; j < HIDc; ++j) m += p[j];
                m *= (1.f / HIDc);
                float v = 0.f;
                for (int j = 0; j < HIDc; ++j) { const float d = p[j] - m; v += d * d; }
                v *= (1.f / HIDc);
                rst[tid * 2] = m; rst[tid * 2 + 1] = rsqrtf(v + EPSc);
            }
            __syncthreads();
            // ---- P6d: h = sig(go)*tanh(LN(c)) ----
            for (int e = tid; e < 16 * HIDc; e += NTH) {
                const int r = e / HIDc, k = e % HIDc;
                const float go = gat[r * 100 + 3 * HIDc + k];
                const float nrm = (cS[r * HIDc + k] - rst[r * 2]) * rst[r * 2 + 1]
                                  * W[W_LCG + k] + W[W_LCB + k];
                hS[r * 28 + k] = sigm(go) * tanhf(nrm);
            }
            __syncthreads();
        } // layers

        // ---- output head: m = h1 @ Wr^T + br ; accumulate sigmoid(m) ----
        if (wave == 0) {
            const float* h1 = S + O_ACT + A_H + 448;
            v8f acc = wmma_np(h1, 28, S + O_WR, 28, 0, v8zero(), lane);
            const int col = lane & 15, rb = (lane >> 4) << 3;
            if (col < MAPc) {
                const float brv = S[O_BR + col];
#pragma unroll
                for (int i = 0; i < 8; ++i) {
                    const int r = rb + i;
                    const float mv = acc[i] + brv;
                    out[OUT_M + ((size_t)(b0 + r) * Tc + t) * MAPc + col] = mv;
                    nac[r * MAPc + col] += sigm(mv);
                }
            }
        }
        __syncthreads();
    } // time

    // ---- epilogue: num + final hidden states ----
    for (int e = tid; e < 16 * MAPc; e += NTH) {
        const int r = e / MAPc, c = e % MAPc;
        out[(size_t)(b0 + r) * MAPc + c] = nac[r * MAPc + c];
    }
    for (int e = tid; e < 2 * 16 * HIDc; e += NTH) {
        const int l = e / 400, rem = e % 400, r = rem / HIDc, k = rem % HIDc;
        const size_t gi = (size_t)l * Bc * HIDc + (size_t)(b0 + r) * HIDc + k;
        out[OUT_H + gi] = S[O_ACT + A_H + l * 448 + r * 28 + k];
        out[OUT_C + gi] = S[O_ACT + A_C + l * 400 + r * HIDc + k];
    }
    for (int e = tid; e < 2 * 16 * HYPc; e += NTH) {
        const int l = e / 416, rem = e % 416, r = rem / HYPc, j = rem % HYPc;
        const size_t gi = (size_t)l * Bc * HYPc + (size_t)(b0 + r) * HYPc + j;
        out[OUT_HH + gi] = S[O_ACT + A_HH + l * 448 + r * 28 + j];
        out[OUT_CH + gi] = S[O_ACT + A_CH + l * 416 + r * HYPc + j];
    }
}

extern "C" void kernel_launch(void* const* d_in, const int* in_sizes, int n_in,
                              void* d_out, int out_size, void* d_ws, size_t ws_size,
                              hipStream_t stream) {
    (void)in_sizes; (void)out_size; (void)d_ws; (void)ws_size;
    Args a;
    for (int i = 0; i < 53; ++i) a.q[i] = (i < n_in) ? (const float*)d_in[i] : nullptr;
    hipFuncSetAttribute(reinterpret_cast<const void*>(hyperlstm_kernel),
                        hipFuncAttributeMaxDynamicSharedMemorySize, (int)LDS_BYTES);
    hyperlstm_kernel<<<Bc / 16, NTH, LDS_BYTES, stream>>>(a, (float*)d_out);
}